// SSMCore_87290915324131
// MI455X (gfx1250) — compile-verified
//
#include <hip/hip_runtime.h>
#include <math.h>

// ---------------- problem constants ----------------
#define BB     2
#define TT     1024
#define DD     2048      // d_inner
#define SS     16        // d_state
#define RR     128       // dt_rank
#define KK     4         // conv kernel
#define NPROJ  160       // dt_rank + 2*d_state
#define NCHUNK 8
#define CHUNK  128       // TT / NCHUNK
#define LOG_EPS  (-87.498233f)   // log(1e-38)
#define NEG_BIG  (-1e30f)

typedef __attribute__((ext_vector_type(2))) float v2f;
typedef __attribute__((ext_vector_type(8))) float v8f;

__device__ __forceinline__ float logaddexp_f(float x, float y) {
    float m = fmaxf(x, y);
    float d = fminf(x, y) - m;          // <= 0; exp underflows cleanly to 0
    return m + log1pf(__expf(d));
}

// ---------------- 1) depthwise causal conv + SiLU ----------------
__global__ __launch_bounds__(256)
void k_conv_silu(const float* __restrict__ x, const float* __restrict__ cw,
                 const float* __restrict__ cb, float* __restrict__ xc) {
    int idx = blockIdx.x * blockDim.x + threadIdx.x;   // (b,t,d) flat, 2^22
    int d = idx & (DD - 1);
    int t = (idx >> 11) & (TT - 1);
    int b = idx >> 21;
    float acc = cb[d];
#pragma unroll
    for (int k = 0; k < KK; ++k) {
        int tt = t - (KK - 1) + k;
        if (tt >= 0) acc += x[(b * TT + tt) * DD + d] * cw[d * KK + k];
    }
    xc[idx] = acc * (1.0f / (1.0f + __expf(-acc)));    // silu
}

// ---------------- 2) proj = xc @ x_proj_w^T  (M=2048,K=2048,N=160), f32 WMMA ----------------
__global__ __launch_bounds__(256)
void k_gemm_proj(const float* __restrict__ A, const float* __restrict__ W,
                 float* __restrict__ proj) {
    int tid  = blockIdx.x * blockDim.x + threadIdx.x;
    int wave = tid >> 5;           // 1280 waves: 128 mTiles x 10 nTiles
    int lane = tid & 31;
    int nT = wave % 10, mT = wave / 10;
    int m0 = mT * 16, n0 = nT * 16;
    int ln  = lane & 15;
    int klo = (lane >> 4) * 2;     // A 16x4: lanes0-15 K=0,1 ; lanes16-31 K=2,3
    const float* aptr = A + (size_t)(m0 + ln) * DD + klo;
    const float* bptr = W + (size_t)(n0 + ln) * DD + klo;   // B[k][n] = W[n*K + k]
    v8f c = {0.f,0.f,0.f,0.f,0.f,0.f,0.f,0.f};
    for (int k0 = 0; k0 < DD; k0 += 4) {
        v2f a = *(const v2f*)(aptr + k0);
        v2f b = *(const v2f*)(bptr + k0);
        c = __builtin_amdgcn_wmma_f32_16x16x4_f32(false, a, false, b,
                                                  (short)0, c, false, false);
    }
#pragma unroll
    for (int i = 0; i < 8; ++i) {
        int row = m0 + i + 8 * (lane >> 4);              // C layout: M = vgpr + 8*(lane/16)
        proj[row * NPROJ + n0 + ln] = c[i];
    }
}

// ---------------- 3) dt = softplus(proj[:, :128] @ dt_proj_w^T + b)  (M=2048,K=128,N=2048) ----------------
__global__ __launch_bounds__(256)
void k_gemm_dt(const float* __restrict__ proj, const float* __restrict__ Wdt,
               const float* __restrict__ bdt, float* __restrict__ dt) {
    int tid  = blockIdx.x * blockDim.x + threadIdx.x;
    int wave = tid >> 5;           // 16384 waves: 128 x 128 tiles
    int lane = tid & 31;
    int nT = wave & 127, mT = wave >> 7;
    int m0 = mT * 16, n0 = nT * 16;
    int ln  = lane & 15;
    int klo = (lane >> 4) * 2;
    const float* aptr = proj + (size_t)(m0 + ln) * NPROJ + klo;   // dt_raw = first 128 cols
    const float* bptr = Wdt  + (size_t)(n0 + ln) * RR    + klo;
    v8f c = {0.f,0.f,0.f,0.f,0.f,0.f,0.f,0.f};
#pragma unroll
    for (int k0 = 0; k0 < RR; k0 += 4) {
        v2f a = *(const v2f*)(aptr + k0);
        v2f b = *(const v2f*)(bptr + k0);
        c = __builtin_amdgcn_wmma_f32_16x16x4_f32(false, a, false, b,
                                                  (short)0, c, false, false);
    }
#pragma unroll
    for (int i = 0; i < 8; ++i) {
        int row = m0 + i + 8 * (lane >> 4);
        int n   = n0 + ln;
        float v = c[i] + bdt[n];
        dt[row * DD + n] = (v > 20.f) ? v : log1pf(__expf(v));    // softplus
    }
}

// ---------------- 4a) per-chunk scan reduction: (Asum, LU) per (b,j,d,s) ----------------
__global__ __launch_bounds__(256)
void k_scan_chunk(const float* __restrict__ dt, const float* __restrict__ xc,
                  const float* __restrict__ proj, const float* __restrict__ A_log,
                  float* __restrict__ Asum, float* __restrict__ LU) {
    int tid = blockIdx.x * blockDim.x + threadIdx.x;   // 2^19 threads
    int s = tid & 15;
    int d = (tid >> 4) & (DD - 1);
    int j = (tid >> 15) & (NCHUNK - 1);
    int b = tid >> 18;
    float a  = -__expf(A_log[d * SS + s]);
    float as = 0.f, lu = NEG_BIG;
    int t0 = j * CHUNK;
    for (int t = t0; t < t0 + CHUNK; ++t) {
        int row   = b * TT + t;
        float dtv = dt[row * DD + d];
        float xv  = xc[row * DD + d];
        float Bv  = proj[row * NPROJ + RR + s];
        float ldA  = fmaxf(dtv * a, LOG_EPS);
        float ldbx = __logf(fmaxf(dtv * Bv * xv, 1e-38f));
        as += ldA;
        lu  = logaddexp_f(lu + ldA, ldbx);   // h_end = exp(as)*h_in + exp(lu)
    }
    int st = ((b * NCHUNK + j) * DD + d) * SS + s;
    Asum[st] = as;
    LU[st]   = lu;
}

// ---------------- 4b) serial combine across chunks -> incoming log-state per chunk ----------------
__global__ __launch_bounds__(256)
void k_scan_combine(const float* __restrict__ Asum, const float* __restrict__ LU,
                    float* __restrict__ carry) {
    int tid = blockIdx.x * blockDim.x + threadIdx.x;   // 2^16 threads: (b,d,s)
    int s = tid & 15;
    int d = (tid >> 4) & (DD - 1);
    int b = tid >> 15;
    float c = NEG_BIG;
#pragma unroll
    for (int j = 0; j < NCHUNK; ++j) {
        int st = ((b * NCHUNK + j) * DD + d) * SS + s;
        carry[st] = c;                                  // log h entering chunk j
        c = logaddexp_f(c + Asum[st], LU[st]);
    }
}

// ---------------- 4c) re-walk chunk with prefix, contract over S, write y ----------------
__global__ __launch_bounds__(256)
void k_scan_apply(const float* __restrict__ dt, const float* __restrict__ xc,
                  const float* __restrict__ proj, const float* __restrict__ A_log,
                  const float* __restrict__ Dp, const float* __restrict__ carry,
                  float* __restrict__ y) {
    int tid  = blockIdx.x * blockDim.x + threadIdx.x;
    int lane = threadIdx.x & 31;
    int wave = tid >> 5;             // 16384 waves: b(1) | j(3) | dpair(10)
    int s  = lane & 15;
    int dh = lane >> 4;              // two d channels per wave
    int dpair = wave & (DD / 2 - 1);
    int j = (wave >> 10) & (NCHUNK - 1);
    int b = wave >> 13;
    int d = dpair * 2 + dh;
    float a  = -__expf(A_log[d * SS + s]);
    float Dv = Dp[d];
    int st = ((b * NCHUNK + j) * DD + d) * SS + s;
    float lh = carry[st];
    int t0 = j * CHUNK;
    for (int t = t0; t < t0 + CHUNK; ++t) {
        int row   = b * TT + t;
        float dtv = dt[row * DD + d];
        float xv  = xc[row * DD + d];
        float Bv  = proj[row * NPROJ + RR + s];
        float Cv  = proj[row * NPROJ + RR + SS + s];
        float ldA  = fmaxf(dtv * a, LOG_EPS);
        float ldbx = __logf(fmaxf(dtv * Bv * xv, 1e-38f));
        lh = logaddexp_f(lh + ldA, ldbx);
        float val = Cv * __expf(lh);
        // reduce over the 16 state lanes (stays inside each 16-lane half)
#pragma unroll
        for (int off = 8; off >= 1; off >>= 1) val += __shfl_xor(val, off, 32);
        if (s == 0) y[row * DD + d] = val + Dv * xv;
    }
}

// ---------------- launcher ----------------
extern "C" void kernel_launch(void* const* d_in, const int* in_sizes, int n_in,
                              void* d_out, int out_size, void* d_ws, size_t ws_size,
                              hipStream_t stream) {
    const float* x        = (const float*)d_in[0];
    const float* conv_w   = (const float*)d_in[1];
    const float* conv_b   = (const float*)d_in[2];
    const float* x_proj_w = (const float*)d_in[3];
    const float* dt_proj_w= (const float*)d_in[4];
    const float* dt_proj_b= (const float*)d_in[5];
    const float* A_log    = (const float*)d_in[6];
    const float* D_param  = (const float*)d_in[7];
    float* y  = (float*)d_out;

    // workspace carve-up (floats): total ~10.3M floats (~41 MB)
    float* ws    = (float*)d_ws;
    float* xc    = ws;                                       // B*T*D      = 4,194,304
    float* proj  = xc   + (size_t)BB * TT * DD;              // B*T*160    =   327,680
    float* dtb   = proj + (size_t)BB * TT * NPROJ;           // B*T*D      = 4,194,304
    float* Asum  = dtb  + (size_t)BB * TT * DD;              // B*NC*D*S   =   524,288
    float* LUb   = Asum + (size_t)BB * NCHUNK * DD * SS;     //            =   524,288
    float* carry = LUb  + (size_t)BB * NCHUNK * DD * SS;     //            =   524,288

    k_conv_silu  <<<(BB * TT * DD) / 256, 256, 0, stream>>>(x, conv_w, conv_b, xc);
    k_gemm_proj  <<<160,  256, 0, stream>>>(xc, x_proj_w, proj);          // 1280 waves
    k_gemm_dt    <<<2048, 256, 0, stream>>>(proj, dt_proj_w, dt_proj_b, dtb); // 16384 waves
    k_scan_chunk <<<2048, 256, 0, stream>>>(dtb, xc, proj, A_log, Asum, LUb);
    k_scan_combine<<<256, 256, 0, stream>>>(Asum, LUb, carry);
    k_scan_apply <<<2048, 256, 0, stream>>>(dtb, xc, proj, A_log, D_param, carry, y);
}